// DPDPQuantizer_3659312136599
// MI455X (gfx1250) — compile-verified
//
#include <hip/hip_runtime.h>

#define T_N 8192
#define D_N 1024
#define K_N 1024
#define LMBDA 5.0f
#define BIG_F 3.0e38f

typedef float v2f __attribute__((ext_vector_type(2)));
typedef float v8f __attribute__((ext_vector_type(8)));

// ---------------- row squared-norms (one block per row) ----------------
__global__ void row_norm_kernel(const float* __restrict__ X, float* __restrict__ out, int ncols) {
  const int row = blockIdx.x;
  const float* xr = X + (size_t)row * ncols;
  float s = 0.f;
  for (int c = threadIdx.x; c < ncols; c += blockDim.x) {
    float v = xr[c];
    s += v * v;
  }
  __shared__ float red[256];
  red[threadIdx.x] = s;
  __syncthreads();
  for (int off = 128; off > 0; off >>= 1) {
    if ((int)threadIdx.x < off) red[threadIdx.x] += red[threadIdx.x + off];
    __syncthreads();
  }
  if (threadIdx.x == 0) out[row] = red[0];
}

// ---- d2[t,k] = fn2[t] + cn2[k] - 2 * F@CB^T  via V_WMMA_F32_16X16X4_F32 ----
// Block = 256 threads (8 waves) arranged 2 row-tiles x 4 col-strips.
// Each wave computes a 16x64 strip (4 tiles): A fragment loaded ONCE per
// k-step and reused across 4 WMMAs -> 1.25 loads per wmma instead of 2.
__global__ void dist_gemm_kernel(const float* __restrict__ F, const float* __restrict__ CB,
                                 const float* __restrict__ fn2, const float* __restrict__ cn2,
                                 float* __restrict__ d2) {
  const int lane = threadIdx.x & 31;
  const int wave = threadIdx.x >> 5;   // 0..7
  const int wrow = wave >> 2;          // 0..1  (row tile within block)
  const int wcol = wave & 3;           // 0..3  (64-col strip within block)
  const int half = lane >> 4;          // A/B frag: lane<16 -> K={0,1}, lane>=16 -> K={2,3}
  const int l16  = lane & 15;

  const int row0 = blockIdx.y * 32 + wrow * 16;
  const int col0 = blockIdx.x * 256 + wcol * 64;

  const float* arow = F  + (size_t)(row0 + l16) * D_N + 2 * half;
  const float* brow = CB + (size_t)(col0 + l16) * D_N + 2 * half;

  v8f acc0 = {}, acc1 = {}, acc2 = {}, acc3 = {};
  #pragma unroll 4
  for (int k0 = 0; k0 < D_N; k0 += 4) {
    v2f a  = *(const v2f*)(arow + k0);
    v2f b0 = *(const v2f*)(brow + k0);
    v2f b1 = *(const v2f*)(brow + 16 * D_N + k0);
    v2f b2 = *(const v2f*)(brow + 32 * D_N + k0);
    v2f b3 = *(const v2f*)(brow + 48 * D_N + k0);
    acc0 = __builtin_amdgcn_wmma_f32_16x16x4_f32(false, a, false, b0, (short)0, acc0, false, false);
    acc1 = __builtin_amdgcn_wmma_f32_16x16x4_f32(false, a, false, b1, (short)0, acc1, false, false);
    acc2 = __builtin_amdgcn_wmma_f32_16x16x4_f32(false, a, false, b2, (short)0, acc2, false, false);
    acc3 = __builtin_amdgcn_wmma_f32_16x16x4_f32(false, a, false, b3, (short)0, acc3, false, false);
  }

  // epilogue: d2 = fn2[row] + cn2[col] - 2*dot   (C/D layout: lane half -> M 0..7 / 8..15)
  const int rb = row0 + half * 8;
  float fr[8];
  #pragma unroll
  for (int j = 0; j < 8; ++j) fr[j] = fn2[rb + j];

  v8f accs[4] = {acc0, acc1, acc2, acc3};
  #pragma unroll
  for (int s = 0; s < 4; ++s) {
    const int col = col0 + 16 * s + l16;
    const float cn = cn2[col];
    #pragma unroll
    for (int j = 0; j < 8; ++j) {
      d2[(size_t)(rb + j) * K_N + col] = fr[j] + cn - 2.0f * accs[s][j];
    }
  }
}

// ---------------- sequential DP forward scan (single WGP) ----------------
// 1024 threads = 32 waves; thread k owns f_k, seg_k in registers.
__global__ void dp_forward_kernel(const float* __restrict__ d2,
                                  int* __restrict__ betas, int* __restrict__ gammas) {
  const int k    = threadIdx.x;          // 0..1023
  const int lane = k & 31;
  const int wv   = k >> 5;
  __shared__ float sf[32];
  __shared__ int   sk[32];
  __shared__ int   ss[32];
  __shared__ float s_alpha;

  float f = BIG_F;
  int   seg = 0;
  float alpha = 0.0f;
  float dcur = d2[k];                    // prefetch row 0

  for (int t = 0; t < T_N; ++t) {
    // prefetch next row early; hides L2 latency behind reductions/barriers
    float dnext = (t + 1 < T_N) ? d2[(size_t)(t + 1) * K_N + k] : 0.0f;

    float ext = f - LMBDA;
    bool take_new = alpha < ext;         // start a new segment at t
    f = dcur + (take_new ? alpha : ext);
    if (take_new) seg = t;

    // argmin over (value, index, seg-start); tie-break to lower index (jnp.argmin)
    float rf = f; int rk = k; int rs = seg;
    #pragma unroll
    for (int off = 16; off > 0; off >>= 1) {
      float of = __shfl_down(rf, off, 32);
      int   ok = __shfl_down(rk, off, 32);
      int   os = __shfl_down(rs, off, 32);
      if (of < rf || (of == rf && ok < rk)) { rf = of; rk = ok; rs = os; }
    }
    if (lane == 0) { sf[wv] = rf; sk[wv] = rk; ss[wv] = rs; }
    __syncthreads();
    if (wv == 0) {
      rf = sf[lane]; rk = sk[lane]; rs = ss[lane];
      #pragma unroll
      for (int off = 16; off > 0; off >>= 1) {
        float of = __shfl_down(rf, off, 32);
        int   ok = __shfl_down(rk, off, 32);
        int   os = __shfl_down(rs, off, 32);
        if (of < rf || (of == rf && ok < rk)) { rf = of; rk = ok; rs = os; }
      }
      if (lane == 0) {
        betas[t]  = rs;                  // segment start of winner
        gammas[t] = rk;                  // winning codeword
        s_alpha   = rf;                  // new alpha
      }
    }
    __syncthreads();
    alpha = s_alpha;
    dcur = dnext;
  }
}

// ---------------- backtrack (LDS-staged pointer chase) ----------------
__global__ void backtrack_kernel(const int* __restrict__ betas, const int* __restrict__ gammas,
                                 int* __restrict__ units, float* __restrict__ units_f) {
  __shared__ int sb[T_N];   // 32KB
  __shared__ int sg[T_N];   // 32KB  (CDNA5: up to 320KB LDS / workgroup)
  for (int i = threadIdx.x; i < T_N; i += blockDim.x) { sb[i] = betas[i]; sg[i] = gammas[i]; }
  __syncthreads();
  if (threadIdx.x == 0) {
    int cur = T_N;
    for (int p = T_N - 1; p >= 0; --p) {
      int g = sg[cur - 1];
      int b = sb[cur - 1];
      units[p]   = g;
      units_f[p] = (float)g;             // tuple output #2 (int -> f32 slot)
      if (p == b) cur = p;
    }
  }
}

// ---------------- gather: quantized[t,:] = codebook[units[t],:] ----------------
__global__ void gather_kernel(const int* __restrict__ units, const float* __restrict__ CB,
                              float* __restrict__ out) {
  const int t = blockIdx.x;
  const int u = units[t];
  const float4* src = (const float4*)(CB + (size_t)u * D_N);
  float4* dst = (float4*)(out + (size_t)t * D_N);
  dst[threadIdx.x] = src[threadIdx.x];   // blockDim = 256 == D/4
}

extern "C" void kernel_launch(void* const* d_in, const int* in_sizes, int n_in,
                              void* d_out, int out_size, void* d_ws, size_t ws_size,
                              hipStream_t stream) {
  const float* features = (const float*)d_in[0];   // (T, D) f32
  const float* codebook = (const float*)d_in[1];   // (K, D) f32
  float* out = (float*)d_out;                      // [T*D quantized | T units-as-f32]

  // workspace layout
  char*  ws     = (char*)d_ws;
  float* d2     = (float*)ws;                                  // T*K f32 = 32MB
  float* fn2    = (float*)(ws + (size_t)T_N * K_N * 4);        // T f32
  float* cn2    = fn2 + T_N;                                   // K f32
  int*   betas  = (int*)(cn2 + K_N);                           // T i32
  int*   gammas = betas + T_N;                                 // T i32
  int*   units  = gammas + T_N;                                // T i32

  row_norm_kernel<<<T_N, 256, 0, stream>>>(features, fn2, D_N);
  row_norm_kernel<<<K_N, 256, 0, stream>>>(codebook, cn2, D_N);

  dim3 ggrid(K_N / 256, T_N / 32);                 // (4, 256)
  dist_gemm_kernel<<<ggrid, 256, 0, stream>>>(features, codebook, fn2, cn2, d2);

  dp_forward_kernel<<<1, 1024, 0, stream>>>(d2, betas, gammas);

  backtrack_kernel<<<1, 256, 0, stream>>>(betas, gammas, units, out + (size_t)T_N * D_N);

  gather_kernel<<<T_N, 256, 0, stream>>>(units, codebook, out);
}